// SelfAttention3D_24575802867792
// MI455X (gfx1250) — compile-verified
//
#include <hip/hip_runtime.h>
#include <math.h>

// ---------------------------------------------------------------------------
// SelfAttention3D for MI455X (gfx1250, wave32, WMMA f32 16x16x4)
//   B=4, C=64, Cqk=8, N=T*H*W=8192
//   Phase 1: qkv projections (token-major layouts in workspace)
//   Phase 2: TWO-PASS flash attention:
//     Pass A: row max over all keys (per-lane running max, no shuffles in loop)
//     Pass B: P = exp(S - m), l = sum(P), O = P*V   (no online rescaling)
//   Each wave owns TWO 16-query tiles so every staged K/V tile + barrier
//   feeds 36 WMMAs instead of 18. K+V per batch = 2.36 MB -> L2-resident.
//   Next tile is prefetched (global_prefetch_b8) while current one computes.
// ---------------------------------------------------------------------------

typedef float v2f __attribute__((ext_vector_type(2)));
typedef float v8f __attribute__((ext_vector_type(8)));

constexpr int Bc   = 4;
constexpr int Cc   = 64;
constexpr int Cqk  = 8;
constexpr int Nn   = 8192;      // tokens per batch
constexpr int KT   = 16;        // keys per tile
constexpr int QT   = 16;        // queries per WMMA tile
constexpr int QTW  = 2;         // query tiles per wave
constexpr int WAVES = 4;        // waves per block

// ---------------------------------------------------------------------------
// Kernel 1: per-token linear projections (1x1 conv3d == channel matvec)
//   x  : [B, C, N]   (channel-major, as in reference)
//   q,k: [B, N, 8]   token-major for WMMA fragment loads
//   v  : [B, N, 64]
// ---------------------------------------------------------------------------
__global__ __launch_bounds__(256)
void qkv_proj_kernel(const float* __restrict__ x,
                     const float* __restrict__ Wq, const float* __restrict__ bq,
                     const float* __restrict__ Wk, const float* __restrict__ bk,
                     const float* __restrict__ Wv, const float* __restrict__ bv,
                     float* __restrict__ q, float* __restrict__ k,
                     float* __restrict__ v)
{
    __shared__ float sWq[Cqk * Cc];
    __shared__ float sWk[Cqk * Cc];
    __shared__ float sWv[Cc * Cc];
    __shared__ float sbq[Cqk], sbk[Cqk], sbv[Cc];

    for (int i = threadIdx.x; i < Cqk * Cc; i += 256) { sWq[i] = Wq[i]; sWk[i] = Wk[i]; }
    for (int i = threadIdx.x; i < Cc * Cc;  i += 256) { sWv[i] = Wv[i]; }
    if (threadIdx.x < Cqk) { sbq[threadIdx.x] = bq[threadIdx.x]; sbk[threadIdx.x] = bk[threadIdx.x]; }
    if (threadIdx.x < Cc)  { sbv[threadIdx.x] = bv[threadIdx.x]; }
    __syncthreads();

    const int t = blockIdx.x * 256 + threadIdx.x;   // global token id in [0, B*N)
    const int b = t / Nn;
    const int n = t % Nn;

    float qa[Cqk], ka[Cqk], va[Cc];
#pragma unroll
    for (int d = 0; d < Cqk; ++d) { qa[d] = sbq[d]; ka[d] = sbk[d]; }
#pragma unroll
    for (int d = 0; d < Cc; ++d)  { va[d] = sbv[d]; }

    const float* xb = x + (size_t)b * Cc * Nn + n;
    for (int c = 0; c < Cc; ++c) {
        const float xv = xb[(size_t)c * Nn];
#pragma unroll
        for (int d = 0; d < Cqk; ++d) {
            qa[d] = fmaf(sWq[d * Cc + c], xv, qa[d]);
            ka[d] = fmaf(sWk[d * Cc + c], xv, ka[d]);
        }
#pragma unroll
        for (int d = 0; d < Cc; ++d) {
            va[d] = fmaf(sWv[d * Cc + c], xv, va[d]);
        }
    }

    float4* qo = (float4*)(q + (size_t)t * Cqk);
    float4* ko = (float4*)(k + (size_t)t * Cqk);
    qo[0] = make_float4(qa[0], qa[1], qa[2], qa[3]);
    qo[1] = make_float4(qa[4], qa[5], qa[6], qa[7]);
    ko[0] = make_float4(ka[0], ka[1], ka[2], ka[3]);
    ko[1] = make_float4(ka[4], ka[5], ka[6], ka[7]);
    float4* vo = (float4*)(v + (size_t)t * Cc);
#pragma unroll
    for (int d = 0; d < Cc / 4; ++d)
        vo[d] = make_float4(va[4 * d], va[4 * d + 1], va[4 * d + 2], va[4 * d + 3]);
}

// ---------------------------------------------------------------------------
// Kernel 2: two-pass flash attention, f32 WMMA, 2 query tiles per wave.
//
// WMMA f32 16x16x4 VGPR layouts (ISA 7.12.2, wave32, lane L = 16*h + r):
//   A (16x4, MxK): a.x = A[r][2h],   a.y = A[r][2h+1]
//   B (4x16, KxN): b.x = B[2h][r],   b.y = B[2h+1][r]
//   C/D (16x16):   d[i] = D[i + 8h][r]
// ---------------------------------------------------------------------------
__global__ __launch_bounds__(WAVES * 32)
void flash_attn_kernel(const float* __restrict__ q, const float* __restrict__ k,
                       const float* __restrict__ v, const float* __restrict__ x,
                       const float* __restrict__ gamma, float* __restrict__ out)
{
    __shared__ float kt_lds[KT][Cqk];               // K tile [key][ch]
    __shared__ float vt_lds[Cc][KT + 2];            // V tile transposed [ch][key], padded
    __shared__ float p_lds[WAVES][QTW][QT][KT + 2]; // per-wave P tiles, padded

    const int tid  = threadIdx.x;
    const int wave = tid >> 5;
    const int lane = tid & 31;
    const int h    = lane >> 4;   // half-wave
    const int r    = lane & 15;   // index within half-wave

    const int blocksPerBatch = Nn / (QT * QTW * WAVES);       // 64
    const int b  = blockIdx.x / blocksPerBatch;
    const int q0 = (blockIdx.x % blocksPerBatch) * (QT * QTW * WAVES)
                 + wave * (QT * QTW);                         // two tiles: q0, q0+16

    const float* qB = q + (size_t)b * Nn * Cqk;
    const float* kB = k + (size_t)b * Nn * Cqk;
    const float* vB = v + (size_t)b * Nn * Cc;

    // Preload Q A-fragments: [query tile][K-chunk]
    v2f qa[QTW][2];
#pragma unroll
    for (int qt = 0; qt < QTW; ++qt) {
        qa[qt][0] = *(const v2f*)(qB + (size_t)(q0 + qt * QT + r) * Cqk + 2 * h);
        qa[qt][1] = *(const v2f*)(qB + (size_t)(q0 + qt * QT + r) * Cqk + 4 + 2 * h);
    }

    // =====================================================================
    // Pass A: global row max.  Per-lane running max only — no cross-lane
    // traffic inside the loop.  K streamed straight from global (L2 hits).
    // =====================================================================
    float mrow[QTW][8];
#pragma unroll
    for (int qt = 0; qt < QTW; ++qt)
#pragma unroll
        for (int i = 0; i < 8; ++i) mrow[qt][i] = -INFINITY;

    for (int kt0 = 0; kt0 < Nn; kt0 += KT) {
        const v2f kb0 = *(const v2f*)(kB + (size_t)(kt0 + r) * Cqk + 2 * h);
        const v2f kb1 = *(const v2f*)(kB + (size_t)(kt0 + r) * Cqk + 4 + 2 * h);
#pragma unroll
        for (int qt = 0; qt < QTW; ++qt) {
            v8f s = v8f{};
            s = __builtin_amdgcn_wmma_f32_16x16x4_f32(false, qa[qt][0], false, kb0,
                                                      (short)0, s, false, false);
            s = __builtin_amdgcn_wmma_f32_16x16x4_f32(false, qa[qt][1], false, kb1,
                                                      (short)0, s, false, false);
#pragma unroll
            for (int rr = 0; rr < 8; ++rr) mrow[qt][rr] = fmaxf(mrow[qt][rr], s[rr]);
        }
    }
    // One-time cross-lane max within each 16-lane half (row = rr + 8h).
#pragma unroll
    for (int qt = 0; qt < QTW; ++qt)
#pragma unroll
        for (int rr = 0; rr < 8; ++rr)
#pragma unroll
            for (int off = 1; off < 16; off <<= 1)
                mrow[qt][rr] = fmaxf(mrow[qt][rr], __shfl_xor(mrow[qt][rr], off, 32));

    // =====================================================================
    // Pass B: with m fixed, P = exp(S - m); l and O accumulate with no
    // rescaling.  K/V tiles staged in LDS, shared by all 4 waves; next
    // tile prefetched while the current one computes.
    // =====================================================================
    v8f oacc[QTW][4] = {{v8f{}, v8f{}, v8f{}, v8f{}},
                        {v8f{}, v8f{}, v8f{}, v8f{}}};
    float lrow[QTW][8];
#pragma unroll
    for (int qt = 0; qt < QTW; ++qt)
#pragma unroll
        for (int i = 0; i < 8; ++i) lrow[qt][i] = 0.f;

    for (int kt0 = 0; kt0 < Nn; kt0 += KT) {
        // ---- cooperative staging of K (16x8) and V (16x64, transposed) ----
        kt_lds[tid >> 3][tid & 7] = kB[(size_t)(kt0 + (tid >> 3)) * Cqk + (tid & 7)];
#pragma unroll
        for (int i = 0; i < 2; ++i) {
            const int flat = (tid + i * 128) * 4;     // 1024 floats total
            const int j = flat >> 6;                  // key row 0..15
            const int c = flat & 63;                  // channel 0..60 step 4
            const float4 vv = *(const float4*)(vB + (size_t)(kt0 + j) * Cc + c);
            vt_lds[c + 0][j] = vv.x;
            vt_lds[c + 1][j] = vv.y;
            vt_lds[c + 2][j] = vv.z;
            vt_lds[c + 3][j] = vv.w;
        }
        // ---- prefetch next K/V tile into cache (global_prefetch_b8) ----
        if (kt0 + KT < Nn) {
            if (tid < 64) {
                // V tile: 4 KB = 64 x 64B lines
                __builtin_prefetch(vB + (size_t)(kt0 + KT) * Cc + tid * 16, 0, 1);
            } else if (tid < 72) {
                // K tile: 512 B = 8 x 64B lines
                __builtin_prefetch(kB + (size_t)(kt0 + KT) * Cqk + (tid - 64) * 16, 0, 1);
            }
        }
        __syncthreads();

        // ---- S = Q * K^T per query tile (K-dim 8 = two WMMA K=4 chunks) ----
        const v2f kb0 = *(const v2f*)&kt_lds[r][2 * h];
        const v2f kb1 = *(const v2f*)&kt_lds[r][4 + 2 * h];
#pragma unroll
        for (int qt = 0; qt < QTW; ++qt) {
            v8f s = v8f{};
            s = __builtin_amdgcn_wmma_f32_16x16x4_f32(false, qa[qt][0], false, kb0,
                                                      (short)0, s, false, false);
            s = __builtin_amdgcn_wmma_f32_16x16x4_f32(false, qa[qt][1], false, kb1,
                                                      (short)0, s, false, false);
            // P = exp(S - m); accumulate l per-lane; relayout P via LDS
#pragma unroll
            for (int rr = 0; rr < 8; ++rr) {
                const float p = __expf(s[rr] - mrow[qt][rr]);  // elem (rr+8h, r)
                lrow[qt][rr] += p;
                p_lds[wave][qt][rr + 8 * h][r] = p;
            }
        }

        // ---- O += P * V : shared B fragments feed both query tiles ----
#pragma unroll
        for (int j0 = 0; j0 < KT; j0 += 4) {
            const v2f pa0 = *(const v2f*)&p_lds[wave][0][r][j0 + 2 * h];
            const v2f pa1 = *(const v2f*)&p_lds[wave][1][r][j0 + 2 * h];
#pragma unroll
            for (int ct = 0; ct < 4; ++ct) {
                const v2f vb = *(const v2f*)&vt_lds[ct * 16 + r][j0 + 2 * h];
                oacc[0][ct] = __builtin_amdgcn_wmma_f32_16x16x4_f32(
                    false, pa0, false, vb, (short)0, oacc[0][ct], false, false);
                oacc[1][ct] = __builtin_amdgcn_wmma_f32_16x16x4_f32(
                    false, pa1, false, vb, (short)0, oacc[1][ct], false, false);
            }
        }
        __syncthreads();   // protect shared K/V tiles before next staging round
    }

    // One-time cross-lane sum within each 16-lane half.
#pragma unroll
    for (int qt = 0; qt < QTW; ++qt)
#pragma unroll
        for (int rr = 0; rr < 8; ++rr)
#pragma unroll
            for (int off = 1; off < 16; off <<= 1)
                lrow[qt][rr] += __shfl_xor(lrow[qt][rr], off, 32);

    // ---- epilogue: out[b][c][n] = gamma * (O/l) + x[b][c][n] ----
    const float g = gamma[0];
    const float* xB = x   + (size_t)b * Cc * Nn;
    float*       oB = out + (size_t)b * Cc * Nn;
#pragma unroll
    for (int qt = 0; qt < QTW; ++qt) {
#pragma unroll
        for (int rr = 0; rr < 8; ++rr) {
            const float inv = 1.f / lrow[qt][rr];
            const int n = q0 + qt * QT + rr + 8 * h;   // token index of this row
#pragma unroll
            for (int ct = 0; ct < 4; ++ct) {
                const int c = ct * 16 + r;             // channel index
                const size_t a = (size_t)c * Nn + n;
                oB[a] = g * (oacc[qt][ct][rr] * inv) + xB[a];
            }
        }
    }
}

// ---------------------------------------------------------------------------
extern "C" void kernel_launch(void* const* d_in, const int* in_sizes, int n_in,
                              void* d_out, int out_size, void* d_ws, size_t ws_size,
                              hipStream_t stream)
{
    const float* x     = (const float*)d_in[0];
    const float* Wq    = (const float*)d_in[1];
    const float* bq    = (const float*)d_in[2];
    const float* Wk    = (const float*)d_in[3];
    const float* bk    = (const float*)d_in[4];
    const float* Wv    = (const float*)d_in[5];
    const float* bv    = (const float*)d_in[6];
    const float* gamma = (const float*)d_in[7];
    float* out = (float*)d_out;

    float* qws = (float*)d_ws;                        // [B, N, 8]
    float* kws = qws + (size_t)Bc * Nn * Cqk;         // [B, N, 8]
    float* vws = kws + (size_t)Bc * Nn * Cqk;         // [B, N, 64]

    qkv_proj_kernel<<<Bc * Nn / 256, 256, 0, stream>>>(
        x, Wq, bq, Wk, bk, Wv, bv, qws, kws, vws);

    flash_attn_kernel<<<Bc * (Nn / (QT * QTW * WAVES)), WAVES * 32, 0, stream>>>(
        qws, kws, vws, x, gamma, out);
}